// DeepPolyReluLayer_2800318677430
// MI455X (gfx1250) — compile-verified
//
#include <hip/hip_runtime.h>

typedef unsigned int u32;
typedef unsigned long long u64;
typedef __attribute__((ext_vector_type(4))) u32 u32x4;
typedef __attribute__((ext_vector_type(8))) u32 u32x8;
typedef __attribute__((ext_vector_type(4))) float f32x4;

// TDM descriptor geometry: one row = 4096 x 8-byte elements = 32 KiB (matches
// the LDS zero tile), iterated 32x with lds_addr_increment=0 and
// global_addr_increment=32KiB  =>  1 MiB of zeros per tensor_store_from_lds.
#define TILE_ELEMS  4096u
#define TDM_ITERS   32u
#define CHUNK_BYTES (32768ull * TDM_ITERS)   /* 1 MiB per workgroup */

// ---------------------------------------------------------------------------
// Kernel 1: zero-fill the 2*N*N weight region via the Tensor Data Mover.
// Each workgroup zeroes a 32 KiB LDS tile, then wave 0 issues a single
// TENSOR_STORE_FROM_LDS descriptor covering its 1 MiB slice.
// NT store hint: the 512 MiB zero stream is ~2.7x the 192 MB L2 -- don't let
// dead zero lines cycle the whole cache. Device scope so DMA completion is
// device-visible before the diagonal-scatter kernel overwrites 2N floats (64 KB) of it.
// ---------------------------------------------------------------------------
__global__ void __launch_bounds__(256) tdm_zero_fill(float* base) {
  __shared__ float zbuf[8192];            // 32 KiB of zeros (TDM source tile)
  f32x4* z4 = (f32x4*)zbuf;
  f32x4 zz = 0.0f;
  for (u32 i = threadIdx.x; i < 2048u; i += 256u) z4[i] = zz;
  __syncthreads();                        // LDS writes visible before DMA issue

  if (threadIdx.x < 32u) {                // wave 0 only (TDM ignores EXEC)
    u64 gaddr = (u64)(size_t)base + (u64)blockIdx.x * CHUNK_BYTES;
    // Low 32 bits of a flat LDS pointer == wave-relative LDS byte offset
    // (ISA 10.2 aperture mapping).
    u32 lds_base = (u32)(size_t)(void*)zbuf;

    // D# group 0 (128b): count=1, lds_addr, global_addr[56:0], type=2.
    u32x4 g0;
    g0[0] = 1u;                                  // count=1, is_restore=0, gather=0
    g0[1] = lds_base;                            // lds_addr (bytes)
    g0[2] = (u32)(gaddr & 0xffffffffu);          // global_addr[31:0]
    g0[3] = (u32)(gaddr >> 32) | (2u << 30);     // global_addr[56:32] | type=2

    // D# group 1 (256b): no multicast, data_size=8B(3), iterate_enable=1,
    // tensor_dim0 = tile_dim0 = 4096 elems, tensor_dim1 = tile_dim1 = 1.
    u32x8 g1;
    g1[0] = (3u << 16) | (1u << 19);             // data_size=3, iterate_enable
    g1[1] = (TILE_ELEMS & 0xffffu) << 16;        // tensor_dim0[15:0] @ bits 63:48
    g1[2] = (TILE_ELEMS >> 16) | (1u << 16);     // tensor_dim0[31:16] | tensor_dim1[15:0]=1
    g1[3] = (TILE_ELEMS << 16);                  // tensor_dim1[31:16]=0 | tile_dim0
    g1[4] = 1u;                                  // tile_dim1=1, tile_dim2=0
    g1[5] = TILE_ELEMS;                          // tensor_dim0_stride[31:0]
    g1[6] = 0u;                                  // stride hi / dim1_stride lo
    g1[7] = 0u;

    // D# group 2 (128b), iterate mode: keep LDS fixed, advance global by
    // one row (32 KiB) per iteration, 32 iterations.
    u32x4 g2;
    g2[0] = 1u;                                  // tensor_dim2
    g2[1] = 0u;                                  // lds_addr_increment = 0
    g2[2] = TILE_ELEMS;                          // global_addr_increment (8B units)
    g2[3] = (TDM_ITERS - 1u) << 16;              // iterate_count (0 => 1x)

    u32x4 g3 = 0u;                               // dim3/dim4 unused

    asm volatile("tensor_store_from_lds %0, %1, %2, %3 th:TH_STORE_NT scope:SCOPE_DEV"
                 :: "s"(g0), "s"(g1), "s"(g2), "s"(g3)
                 : "memory");
    __builtin_amdgcn_s_wait_tensorcnt(0);        // keep LDS alive until drained
  }
}

// ---------------------------------------------------------------------------
// Fallback tail fill (unused when the region is a multiple of 1 MiB; kept for
// generality). Non-temporal b128 stores: 512 MB >> 192 MB L2, don't thrash it.
// ---------------------------------------------------------------------------
__global__ void zero_tail(f32x4* dst, size_t n4) {
  size_t i = (size_t)blockIdx.x * blockDim.x + threadIdx.x;
  size_t stride = (size_t)gridDim.x * blockDim.x;
  f32x4 z = 0.0f;
  for (; i < n4; i += stride) __builtin_nontemporal_store(z, dst + i);
}

// ---------------------------------------------------------------------------
// Kernel 2: per-neuron three-branch relaxation + diagonal scatter.
// Runs after the fill (same stream => ordered).
// Output layout (flat, return order):
//   [0,N)               x_out
//   [N,2N)              lower_ret
//   [2N,3N)             upper_ret
//   [3N, 3N+N^2)        lower_weights (diag)
//   [3N+N^2, 3N+2N^2)   upper_weights (diag)
//   [3N+2N^2, 4N+2N^2)  lower_bias
//   [4N+2N^2, 5N+2N^2)  upper_bias
// ---------------------------------------------------------------------------
__global__ void __launch_bounds__(256) deeppoly_relu_relax(
    const float* __restrict__ x, const float* __restrict__ lo,
    const float* __restrict__ up, float* __restrict__ out, int N) {
  int i = blockIdx.x * blockDim.x + threadIdx.x;
  if (i >= N) return;

  float xv = x[i], l = lo[i], u = up[i];
  bool neg   = (u <= 0.0f);
  bool pos   = (l >= 0.0f);
  bool cross = !neg && !pos;

  float denom = cross ? (u - l) : 1.0f;
  float slope = cross ? (u / denom) : 0.0f;

  float x_out     = fmaxf(xv, 0.0f);
  float lower_ret = neg ? 0.0f : l;
  float upper_ret = neg ? 0.0f : (pos ? u : slope * u);
  float lw_diag   = neg ? 0.0f : 1.0f;
  float uw_diag   = neg ? 0.0f : (pos ? 1.0f : slope);
  float upper_b   = cross ? slope * l : 0.0f;

  size_t Ns = (size_t)N;
  size_t NN = Ns * Ns;
  out[i]               = x_out;
  out[Ns + i]          = lower_ret;
  out[2 * Ns + i]      = upper_ret;
  out[3 * Ns + (size_t)i * Ns + i]      = lw_diag;   // lower_weights diag
  out[3 * Ns + NN + (size_t)i * Ns + i] = uw_diag;   // upper_weights diag
  out[3 * Ns + 2 * NN + i] = 0.0f;                   // lower_bias
  out[4 * Ns + 2 * NN + i] = upper_b;                // upper_bias
}

extern "C" void kernel_launch(void* const* d_in, const int* in_sizes, int n_in,
                              void* d_out, int out_size, void* d_ws, size_t ws_size,
                              hipStream_t stream) {
  (void)n_in; (void)out_size; (void)d_ws; (void)ws_size;
  const float* x  = (const float*)d_in[0];
  const float* lo = (const float*)d_in[1];
  const float* up = (const float*)d_in[2];
  const size_t N  = (size_t)in_sizes[0];

  float* out   = (float*)d_out;
  float* wbase = out + 3 * N;                       // start of the N^2 blocks
  const u64 fill_bytes = 2ull * N * N * sizeof(float);
  const u64 chunks     = fill_bytes / CHUNK_BYTES;  // N=8192 -> 512 exact
  const u64 tail_bytes = fill_bytes - chunks * CHUNK_BYTES;

  if (chunks)
    hipLaunchKernelGGL(tdm_zero_fill, dim3((u32)chunks), dim3(256), 0, stream,
                       wbase);
  if (tail_bytes) {
    f32x4* tdst = (f32x4*)((char*)wbase + chunks * CHUNK_BYTES);
    hipLaunchKernelGGL(zero_tail, dim3(256), dim3(256), 0, stream, tdst,
                       (size_t)(tail_bytes / 16));
  }
  hipLaunchKernelGGL(deeppoly_relu_relax, dim3((u32)((N + 255) / 256)),
                     dim3(256), 0, stream, x, lo, up, out, (int)N);
}